// SparseDiffDMC_45045617000961
// MI455X (gfx1250) — compile-verified
//
#include <hip/hip_runtime.h>
#include <hip/hip_bf16.h>
#include <math.h>

// ---------------------------------------------------------------------------
// SparseDiffDMC for MI455X (gfx1250, wave32).
// QEF normal equations via V_WMMA_F32_16X16X4_F32 (full fp32 WMMA),
// 4 voxels packed block-diagonally per wave-level matrix op.
// Branchless LDS marshalling AND branchless diagonal-block extraction.
// Edge/corner weights use the gfx1250 hardware V_TANH_F32 trans op.
// ---------------------------------------------------------------------------

#define EPSF      1e-8f
#define WSCALE    0.99f
#define QREG      0.001f

typedef float v2f __attribute__((ext_vector_type(2)));
typedef float v8f __attribute__((ext_vector_type(8)));

__device__ __constant__ int CE[24] = {
    0,1, 1,5, 4,5, 0,4, 2,3, 3,7, 6,7, 2,6, 2,0, 3,1, 7,5, 6,4
};
__device__ __constant__ int CORN[8][3] = {
    {0,0,0},{1,0,0},{0,1,0},{1,1,0},{0,0,1},{1,0,1},{0,1,1},{1,1,1}
};

// gfx1250 hardware tanh (TRANS32). v_nop covers the trans-op result hazard.
__device__ __forceinline__ float htanh(float x) {
    float r;
    asm volatile("v_tanh_f32 %0, %1\n\tv_nop" : "=v"(r) : "v"(x));
    return r;
}

// K1: world[m] = (0 + 0.5)*s - 1 + deform[m]   (the "never scattered" default)
__global__ void dmc_init_world(const float* __restrict__ deform,
                               float* __restrict__ world,
                               int M, const int* __restrict__ resPtr) {
    int i = blockIdx.x * blockDim.x + threadIdx.x;
    if (i >= M) return;
    float s = 2.0f / (float)(*resPtr);
    float base = 0.5f * s - 1.0f;
    world[3*i+0] = base + deform[3*i+0];
    world[3*i+1] = base + deform[3*i+1];
    world[3*i+2] = base + deform[3*i+2];
}

// K2: scatter corner positions (duplicate indices race, matching jax scatter-set)
__global__ void dmc_scatter_world(const int* __restrict__ vox,
                                  const int* __restrict__ cidx,
                                  const float* __restrict__ deform,
                                  float* __restrict__ world,
                                  int N, const int* __restrict__ resPtr) {
    int t = blockIdx.x * blockDim.x + threadIdx.x;
    if (t >= N * 8) return;
    int n = t >> 3, c = t & 7;
    int m = cidx[t];
    float s = 2.0f / (float)(*resPtr);
#pragma unroll
    for (int d = 0; d < 3; ++d) {
        float up = (float)(vox[3*n+d] + CORN[c][d]);
        world[3*m+d] = (up + 0.5f) * s - 1.0f + deform[3*m+d];
    }
}

// K3: main per-voxel pipeline + WMMA QEF
__global__ __launch_bounds__(128) void dmc_main(
    const int*   __restrict__ cidx,
    const float* __restrict__ sdf,
    const float* __restrict__ beta,
    const float* __restrict__ alpha,
    const float* __restrict__ world,
    float*       __restrict__ out,
    int N)
{
    const int tid  = threadIdx.x;        // 0..127
    const int lane = tid & 31;
    const int wave = tid >> 5;           // 0..3
    const int v    = blockIdx.x * 128 + tid;
    const int vc   = (v < N) ? v : (N - 1);   // clamp: keep all lanes executing
    const bool active = (v < N);

    // LDS staging for WMMA marshalling (branchless layouts):
    //   s_a[t][e][i] = wm * nrm_i  (i<3), s_a[t][e][3] = 0   -> A fragment
    //   s_b[t][e][j] = nrm_j       (j<3), s_b[t][e][3] = n.u -> B fragment
    __shared__ float s_a[128][12][4];
    __shared__ float s_b[128][12][4];
    // per wave: 32 voxels x (3 rows x 4 cols), row 32 = dummy sink
    __shared__ __align__(16) float s_ata[4][33][12];

    // ---------------- phase 1: per-lane edge pipeline ----------------
    int ci[8];
    {
        const int4* c4 = (const int4*)(cidx + 8 * vc);      // 32B aligned
        int4 ca = c4[0], cb = c4[1];
        ci[0]=ca.x; ci[1]=ca.y; ci[2]=ca.z; ci[3]=ca.w;
        ci[4]=cb.x; ci[5]=cb.y; ci[6]=cb.z; ci[7]=cb.w;
    }
#pragma unroll
    for (int c = 0; c < 8; ++c)
        __builtin_prefetch(&world[3*ci[c]], 0, 0);          // global_prefetch_b8

    float sv[8], aw[8], bw[12];
    {
        const float4* a4 = (const float4*)(alpha + 8 * vc); // 32B aligned
        float4 aa = a4[0], ab = a4[1];
        float at[8] = {aa.x,aa.y,aa.z,aa.w,ab.x,ab.y,ab.z,ab.w};
#pragma unroll
        for (int c = 0; c < 8; ++c) {
            sv[c] = sdf[ci[c]];
            aw[c] = htanh(at[c]) * WSCALE + 1.0f;
        }
        const float4* b4 = (const float4*)(beta + 12 * vc); // 16B aligned
        float4 b0 = b4[0], b1 = b4[1], b2 = b4[2];
        float bt[12] = {b0.x,b0.y,b0.z,b0.w,b1.x,b1.y,b1.z,b1.w,b2.x,b2.y,b2.z,b2.w};
#pragma unroll
        for (int e = 0; e < 12; ++e)
            bw[e] = htanh(bt[e]) * WSCALE + 1.0f;
    }

    int nocc = 0;
#pragma unroll
    for (int c = 0; c < 8; ++c) nocc += (sv[c] < 0.0f) ? 1 : 0;
    const bool surf = (nocc > 0) && (nocc < 8);

    float wpos[8][3];
#pragma unroll
    for (int c = 0; c < 8; ++c) {
        wpos[c][0] = world[3*ci[c]+0];
        wpos[c][1] = world[3*ci[c]+1];
        wpos[c][2] = world[3*ci[c]+2];
    }

    float ue[12][3], wmv[12];
    float wsum = EPSF;
    float pa[3] = {0,0,0}, na[3] = {0,0,0};

#pragma unroll
    for (int e = 0; e < 12; ++e) {
        const int c0 = CE[2*e], c1 = CE[2*e+1];
        const float s0 = sv[c0], s1 = sv[c1];
        const bool  cross = ((s0 < 0.0f) != (s1 < 0.0f)) && surf;
        const float wm = cross ? 1.0f : 0.0f;
        wmv[e] = wm;
        float denom = s0 - s1;
        float dd = (fabsf(denom) > EPSF) ? denom : EPSF;
        float t  = fminf(fmaxf(s0 / dd, 0.0f), 1.0f);
        float a0 = aw[c0], a1 = aw[c1];
        float ta = a1 * t / (a0 * (1.0f - t) + a1 * t + EPSF);

        float ev[3];
#pragma unroll
        for (int d = 0; d < 3; ++d) ev[d] = wpos[c1][d] - wpos[c0][d];
        float el = sqrtf(ev[0]*ev[0] + ev[1]*ev[1] + ev[2]*ev[2]) + EPSF;
        float dsg = s1 - s0;
        float sg = (dsg > 0.0f) ? 1.0f : ((dsg < 0.0f) ? -1.0f : 0.0f);
        float nr[3];
#pragma unroll
        for (int d = 0; d < 3; ++d) nr[d] = ev[d] / el * sg;
        float nl = sqrtf(nr[0]*nr[0] + nr[1]*nr[1] + nr[2]*nr[2]) + EPSF;
        float dnu = 0.0f;
        const float w = bw[e] * wm;
        wsum += w;
#pragma unroll
        for (int d = 0; d < 3; ++d) {
            nr[d] /= nl;
            ue[e][d] = wpos[c0][d] + ta * ev[d];
            pa[d] += ue[e][d] * w;
            na[d] += nr[d] * w;
            dnu   += nr[d] * ue[e][d];
            s_a[tid][e][d] = nr[d] * wm;
            s_b[tid][e][d] = nr[d];
        }
        s_a[tid][e][3] = 0.0f;     // structural zero row of A
        s_b[tid][e][3] = dnu;      // raw n.u (wm folded in on the A side)
    }
#pragma unroll
    for (int d = 0; d < 3; ++d) { pa[d] /= wsum; na[d] /= wsum; }
    {
        float nal = sqrtf(na[0]*na[0] + na[1]*na[1] + na[2]*na[2]) + EPSF;
#pragma unroll
        for (int d = 0; d < 3; ++d) na[d] /= nal;
    }

    __syncthreads();

    // ---------------- phase 2: WMMA QEF accumulation ----------------
    // [AtA | Atb] = (wm*N)^T (12x3) x [N | n.u] (12x4), 4 voxels
    // block-diagonally per 16x16 fp32 tile, K accumulated in chunks of 4 edges.
    const int m16  = lane & 15;         // A row / B col index
    const int kb   = (lane >> 4) << 1;  // lanes 0-15 -> K 0,1 ; 16-31 -> K 2,3
    const int vloc = m16 >> 2;          // voxel block within group of 4
    const int ic   = m16 & 3;           // component (3 = zero row / n.u col)

    // Branchless extraction setup. Lane holds D[mm][nn] with nn=lane&15,
    // mm=r+8*(lane>>4). Diagonal block b=nn>>2: rows 4b+i live here iff
    // (b>>1)==(lane>>4), at r=4*(b&1)+i. Invalid lanes write to dummy row 32.
    const bool half_ok = ((vloc >> 1) == (lane >> 4));
    const bool hi      = (vloc & 1);

    for (int g = 0; g < 8; ++g) {
        const int vi = wave * 32 + g * 4 + vloc;   // LDS row of this lane's data
        v2f af[3], bf[3];
#pragma unroll
        for (int kc = 0; kc < 3; ++kc) {           // all 12 loads, no branches
            const int e0 = 4*kc + kb;
            af[kc].x = s_a[vi][e0    ][ic];
            af[kc].y = s_a[vi][e0 + 1][ic];
            bf[kc].x = s_b[vi][e0    ][ic];
            bf[kc].y = s_b[vi][e0 + 1][ic];
        }
        v8f acc = {0,0,0,0,0,0,0,0};
        acc = __builtin_amdgcn_wmma_f32_16x16x4_f32(false, af[0], false, bf[0],
                                                    (short)0, acc, false, false);
        acc = __builtin_amdgcn_wmma_f32_16x16x4_f32(false, af[1], false, bf[1],
                                                    (short)0, acc, false, false);
        acc = __builtin_amdgcn_wmma_f32_16x16x4_f32(false, af[2], false, bf[2],
                                                    (short)0, acc, false, false);

        const int row = half_ok ? (g * 4 + vloc) : 32;
        float* dst = &s_ata[wave][row][ic];        // ic == col within block
        dst[0] = hi ? acc[4] : acc[0];
        dst[4] = hi ? acc[5] : acc[1];
        dst[8] = hi ? acc[6] : acc[2];
    }
    asm volatile("s_wait_dscnt 0" ::: "memory");

    float ata[12];                      // 3 rows x 4 cols for this lane's voxel
    {
        const float4* pr = (const float4*)&s_ata[wave][lane][0];
        float4 r0 = pr[0], r1 = pr[1], r2 = pr[2];
        ata[0]=r0.x; ata[1]=r0.y; ata[2]=r0.z; ata[3]=r0.w;
        ata[4]=r1.x; ata[5]=r1.y; ata[6]=r1.z; ata[7]=r1.w;
        ata[8]=r2.x; ata[9]=r2.y; ata[10]=r2.z; ata[11]=r2.w;
    }

    // ---------------- phase 3: solve + L_dev + stores ----------------
    const float A00 = ata[0] + QREG, A01 = ata[1],        A02 = ata[2];
    const float A10 = ata[4],        A11 = ata[5] + QREG, A12 = ata[6];
    const float A20 = ata[8],        A21 = ata[9],        A22 = ata[10] + QREG;
    const float b0 = ata[3]  + QREG * pa[0];
    const float b1 = ata[7]  + QREG * pa[1];
    const float b2 = ata[11] + QREG * pa[2];
    const float det = A00*(A11*A22 - A12*A21)
                    - A01*(A10*A22 - A12*A20)
                    + A02*(A10*A21 - A11*A20);
    const float inv = 1.0f / det;
    float vd[3];
    vd[0] = (b0*(A11*A22-A12*A21) - A01*(b1*A22-A12*b2) + A02*(b1*A21-A11*b2)) * inv;
    vd[1] = (A00*(b1*A22-A12*b2) - b0*(A10*A22-A12*A20) + A02*(A10*b2-b1*A20)) * inv;
    vd[2] = (A00*(A11*b2-b1*A21) - A01*(A10*b2-b1*A20) + b0*(A10*A21-A11*A20)) * inv;

    float dists[12];
    float cnt = EPSF, dsum = 0.0f;
#pragma unroll
    for (int e = 0; e < 12; ++e) {
        float dx = ue[e][0]-pa[0], dy = ue[e][1]-pa[1], dz = ue[e][2]-pa[2];
        dists[e] = sqrtf(dx*dx + dy*dy + dz*dz) * wmv[e];
        dsum += dists[e];
        cnt  += wmv[e];
    }
    const float meanl = dsum / cnt;
    float madsum = 0.0f;
#pragma unroll
    for (int e = 0; e < 12; ++e)
        madsum += fabsf(dists[e] - meanl) * wmv[e];
    const float mask = surf ? 1.0f : 0.0f;
    const float Ldev = (madsum / cnt) * mask;

    if (active) {
        const long nL = (long)N;
#pragma unroll
        for (int d = 0; d < 3; ++d) {
            out[0*nL      + 3*v + d] = pa[d] * mask;   // vertices (== p_alpha*mask)
            out[3*nL      + 3*v + d] = vd[d] * mask;   // vd_qef * mask
            out[7*nL      + 3*v + d] = pa[d] * mask;   // p_alpha * mask
            out[10*nL     + 3*v + d] = na[d] * mask;   // n_alpha * mask
        }
        out[6*nL + v] = Ldev;                          // L_dev
    }
}

extern "C" void kernel_launch(void* const* d_in, const int* in_sizes, int n_in,
                              void* d_out, int out_size, void* d_ws, size_t ws_size,
                              hipStream_t stream) {
    const int*   vox    = (const int*)  d_in[0];   // (N,3) int32
    const float* sdf    = (const float*)d_in[1];   // (M,)  f32
    const int*   cidx   = (const int*)  d_in[2];   // (N,8) int32
    const int*   resPtr = (const int*)  d_in[3];   // scalar int (device)
    const float* deform = (const float*)d_in[4];   // (M,3) f32
    const float* beta   = (const float*)d_in[5];   // (N,12) f32
    const float* alpha  = (const float*)d_in[6];   // (N,8)  f32
    // d_in[7] = gamma : dead code in the reference, unused.

    const int N = in_sizes[0] / 3;
    const int M = in_sizes[1];
    float* world = (float*)d_ws;                   // M*3 floats scratch
    float* out   = (float*)d_out;                  // 13*N floats

    dmc_init_world<<<(M + 255) / 256, 256, 0, stream>>>(deform, world, M, resPtr);
    dmc_scatter_world<<<(N * 8 + 255) / 256, 256, 0, stream>>>(vox, cidx, deform,
                                                               world, N, resPtr);
    dmc_main<<<(N + 127) / 128, 128, 0, stream>>>(cidx, sdf, beta, alpha,
                                                  world, out, N);
}